// CachedE8Quantizer_43224550867084
// MI455X (gfx1250) — compile-verified
//
#include <hip/hip_runtime.h>

// E8 lattice nearest-point quantization, 8M points x 8 fp32.
// Roofline: 512 MiB HBM traffic @ 23.3 TB/s ~= 23 us floor; VALU cost per point
// is the same order, so the kernel minimizes per-point instruction count:
//  - second coset (D8 + 1/2) derived algebraically from the first rounding
//    (f2 = f + 0.5*sgn, d2 = d - 0.5*sgn, flip dir = -sgn) -> no second rndne
//  - math arranged over float2 so the backend can form v_pk_add_f32/v_pk_fma_f32
//  - NT load/store hints (streams exceed the 192 MB L2) + global_prefetch_b8.
// No WMMA: only per-lane 8-wide reductions exist; the matrix pipe would waste
// >=16x of its MACs on off-diagonal terms plus layout shuffles.

typedef float v2f __attribute__((ext_vector_type(2)));
typedef float v4f __attribute__((ext_vector_type(4)));

#ifndef PF_DIST_POINTS
#define PF_DIST_POINTS 2048   // prefetch ~64KB ahead of the wave's stream position
#endif

__global__ __launch_bounds__(256) void e8_quantize_kernel(const float* __restrict__ x,
                                                          float* __restrict__ out,
                                                          int npts) {
  const int idx = blockIdx.x * blockDim.x + threadIdx.x;
  if (idx >= npts) return;

  const v4f* __restrict__ src = (const v4f*)x + (size_t)idx * 2;

  if (idx + PF_DIST_POINTS < npts) {
    __builtin_prefetch((const char*)(src + (size_t)PF_DIST_POINTS * 2), 0, 0);
  }

  const v4f a = __builtin_nontemporal_load(src);
  const v4f b = __builtin_nontemporal_load(src + 1);

  v2f w[4];
  w[0].x = a.x; w[0].y = a.y;
  w[1].x = a.z; w[1].y = a.w;
  w[2].x = b.x; w[2].y = b.y;
  w[3].x = b.z; w[3].y = b.w;

  const v2f vph = {0.5f, 0.5f};
  const v2f vmh = {-0.5f, -0.5f};

  // ---- shared rounding for both cosets ----
  v2f f[4], d[4], sg[4], f2[4], d2[4];
#pragma unroll
  for (int j = 0; j < 4; ++j) {
    f[j].x = rintf(w[j].x);            // v_rndne_f32 == jnp.round (half-to-even)
    f[j].y = rintf(w[j].y);
    d[j] = w[j] - f[j];                // residual, |d| <= 0.5  (pk_add w/ neg)
    sg[j].x = __builtin_copysignf(1.0f, d[j].x);   // v_bfi_b32; d is never -0
    sg[j].y = __builtin_copysignf(1.0f, d[j].y);
    f2[j] = __builtin_elementwise_fma(sg[j], vph, f[j]);   // nearest half-odd pt
    d2[j] = __builtin_elementwise_fma(sg[j], vmh, d[j]);   // residual vs coset 1
  }

  // ---- parities of the (integer-valued) candidate coordinate sums ----
  const v2f sfv = (f[0] + f[1]) + (f[2] + f[3]);
  const v2f sgv = (sg[0] + sg[1]) + (sg[2] + sg[3]);
  const float sumf = sfv.x + sfv.y;                   // = sum(round(x))
  const float sum2 = fmaf(0.5f, sgv.x + sgv.y, sumf); // = sum(f2), integer-valued
  const bool odd0 = (((int)sumf) & 1) != 0;
  const bool odd1 = (((int)sum2) & 1) != 0;           // == parity of sum(round(x-.5))

  // ---- max |residual| per coset (abs = free source modifier) ----
  const float m0 = fmaxf(fmaxf(fmaxf(fabsf(d[0].x), fabsf(d[0].y)),
                               fmaxf(fabsf(d[1].x), fabsf(d[1].y))),
                         fmaxf(fmaxf(fabsf(d[2].x), fabsf(d[2].y)),
                               fmaxf(fabsf(d[3].x), fabsf(d[3].y))));
  const float m1 = fmaxf(fmaxf(fmaxf(fabsf(d2[0].x), fabsf(d2[0].y)),
                               fmaxf(fabsf(d2[1].x), fabsf(d2[1].y))),
                         fmaxf(fmaxf(fabsf(d2[2].x), fabsf(d2[2].y)),
                               fmaxf(fabsf(d2[3].x), fabsf(d2[3].y))));

  // Parity-gated flip thresholds: even parity -> impossible value, no flip.
  const float mth0 = odd0 ? m0 : -1.0f;
  const float mth1 = odd1 ? m1 : -1.0f;

  // ---- build both candidates (flip the max-residual coord toward x) ----
  v2f y0[4], y1[4];
#pragma unroll
  for (int j = 0; j < 4; ++j) {
    v2f a0, a1;
    a0.x = (fabsf(d[j].x) == mth0) ? sg[j].x : 0.0f;
    a0.y = (fabsf(d[j].y) == mth0) ? sg[j].y : 0.0f;
    a1.x = (fabsf(d2[j].x) == mth1) ? -sg[j].x : 0.0f;  // coset-1 flip dir = -sgn
    a1.y = (fabsf(d2[j].y) == mth1) ? -sg[j].y : 0.0f;
    y0[j] = f[j] + a0;     // pk_add
    y1[j] = f2[j] + a1;    // pk_add
  }

  // ---- squared distances (same structure as reference: sum (x-y)^2) ----
  v2f q0 = {0.0f, 0.0f}, q1 = {0.0f, 0.0f};
#pragma unroll
  for (int j = 0; j < 4; ++j) {
    const v2f r0 = w[j] - y0[j];
    const v2f r1 = w[j] - y1[j];
    q0 = __builtin_elementwise_fma(r0, r0, q0);   // pk_fma
    q1 = __builtin_elementwise_fma(r1, r1, q1);   // pk_fma
  }
  const float dist0 = q0.x + q0.y;
  const float dist1 = q1.x + q1.y;

  const bool pick0 = (dist0 <= dist1);
  v4f ra, rb;
  ra.x = pick0 ? y0[0].x : y1[0].x;
  ra.y = pick0 ? y0[0].y : y1[0].y;
  ra.z = pick0 ? y0[1].x : y1[1].x;
  ra.w = pick0 ? y0[1].y : y1[1].y;
  rb.x = pick0 ? y0[2].x : y1[2].x;
  rb.y = pick0 ? y0[2].y : y1[2].y;
  rb.z = pick0 ? y0[3].x : y1[3].x;
  rb.w = pick0 ? y0[3].y : y1[3].y;

  v4f* __restrict__ dst = (v4f*)out + (size_t)idx * 2;
  __builtin_nontemporal_store(ra, dst);
  __builtin_nontemporal_store(rb, dst + 1);
}

extern "C" void kernel_launch(void* const* d_in, const int* in_sizes, int n_in,
                              void* d_out, int out_size, void* d_ws, size_t ws_size,
                              hipStream_t stream) {
  (void)n_in; (void)d_ws; (void)ws_size; (void)out_size;
  const float* x = (const float*)d_in[0];
  float* out = (float*)d_out;
  const int npts = in_sizes[0] / 8;   // 8 fp32 per E8 point

  const int block = 256;              // 8 wave32 waves per block
  const int grid = (npts + block - 1) / block;
  e8_quantize_kernel<<<grid, block, 0, stream>>>(x, out, npts);
}